// MaxPoolingLayer_28424093564963
// MI455X (gfx1250) — compile-verified
//
#include <hip/hip_runtime.h>
#include <hip/hip_bf16.h>
#include <stdint.h>

// Problem constants (from reference): N_NODES=100000, N_EDGES=1600000, D_FEAT=64
#define D_FEAT        64
#define WAVES_TOTAL   8192          // ~196 contiguous edges per wave
#define BLOCK_THREADS 256           // 8 wave32 per block
#define NEG_INF_BITS  0xFF800000u   // float -inf

// Native clang vector type (accepted by nontemporal builtins)
typedef float f32x2 __attribute__((ext_vector_type(2)));

// CDNA5 native float max atomic, no return (STOREcnt path), device scope.
__device__ __forceinline__ void atomic_max_f32(float* addr, float v) {
    asm volatile("global_atomic_max_num_f32 %0, %1, off scope:SCOPE_DEV"
                 :: "v"(addr), "v"(v)
                 : "memory");
}

// Pass 1: seed output with -inf so negative maxima are preserved.
__global__ void __launch_bounds__(BLOCK_THREADS)
seg_init_kernel(uint32_t* __restrict__ out, int n4) {
    int i = blockIdx.x * blockDim.x + threadIdx.x;
    if (i < n4) {
        uint4 v = {NEG_INF_BITS, NEG_INF_BITS, NEG_INF_BITS, NEG_INF_BITS};
        ((uint4*)out)[i] = v;  // global_store_b128
    }
}

// Per-edge body: s/idx are wave-uniform SGPR values (from v_readlane), so the
// segment-change test is a scalar branch and the row load uses saddr form.
__device__ __forceinline__ void process_edge(
    int s, int idx, int& cur, bool& first_seg,
    float& rx, float& ry,
    const float* __restrict__ x, float* __restrict__ out, int foff)
{
    const float NEG_INF = __uint_as_float(NEG_INF_BITS);
    if (s != cur) {
        float* dst = out + (size_t)cur * D_FEAT + foff;
        if (first_seg) {
            // Segment may straddle into the previous wave's range -> atomic.
            atomic_max_f32(dst + 0, rx);
            atomic_max_f32(dst + 1, ry);
            first_seg = false;
        } else {
            // Both boundaries seen inside our range: exclusively owned.
            f32x2 v; v.x = rx; v.y = ry;
            __builtin_nontemporal_store(v, (f32x2*)dst);
        }
        rx = NEG_INF; ry = NEG_INF;
        cur = s;
    }
    // Vectorized gather: global_load_b64 (saddr = row base, voffset = foff).
    const f32x2 v = *(const f32x2*)(x + (size_t)(uint32_t)idx * D_FEAT + foff);
    rx = fmaxf(rx, v.x);
    ry = fmaxf(ry, v.y);
}

// Pass 2: one wave per contiguous edge range. Edges handled in chunks of 32:
// lane l loads edge (base+l)'s seg/nbr once (coalesced b32), values are then
// broadcast per edge via v_readlane. Software-pipelined by one chunk with
// global_prefetch_b8 covering the next chunk's 32 gather rows.
__global__ void __launch_bounds__(BLOCK_THREADS)
seg_max_gather_kernel(const float* __restrict__ x,
                      const int*   __restrict__ nbr,
                      const int*   __restrict__ seg,
                      float*       __restrict__ out,
                      int n_edges, int edges_per_wave)
{
    const int gtid = blockIdx.x * blockDim.x + threadIdx.x;
    const int wave = gtid >> 5;           // wave32
    const int lane = gtid & 31;
    const int foff = lane * 2;            // 2 f32 features per lane (64 total)

    long start = (long)wave * edges_per_wave;
    if (start >= n_edges) return;
    long end = start + edges_per_wave;
    if (end > n_edges) end = n_edges;

    const float NEG_INF = __uint_as_float(NEG_INF_BITS);
    float rx = NEG_INF, ry = NEG_INF;
    int  cur = seg[start];
    bool first_seg = true;  // first segment may straddle into prior wave's range

    // Preload chunk 0 (coalesced: lane l holds edge start+l).
    int sL = 0, idxL = 0;
    {
        long eL = start + lane;
        if (eL < end) {
            sL   = __builtin_nontemporal_load(seg + eL);
            idxL = __builtin_nontemporal_load(nbr + eL);
        }
    }

    for (long base = start; base < end; base += 32) {
        const long rem = end - base;
        const int  cnt = (int)(rem < 32 ? rem : 32);

        // Pipeline: fetch next chunk's edge data; one prefetch instruction
        // covers 32 distinct x rows (per-lane addresses), 2x for the 256B row.
        int sN = 0, idxN = 0;
        {
            long eL = base + 32 + lane;
            if (eL < end) {
                sN   = __builtin_nontemporal_load(seg + eL);
                idxN = __builtin_nontemporal_load(nbr + eL);
                const float* row = x + (size_t)(uint32_t)idxN * D_FEAT;
                __builtin_prefetch(row,      0, 3);   // global_prefetch_b8
                __builtin_prefetch(row + 32, 0, 3);   // second 128B of row
            }
        }

        if (cnt == 32) {
            #pragma unroll 8
            for (int i = 0; i < 32; ++i) {
                const int s   = __builtin_amdgcn_readlane(sL, i);
                const int idx = __builtin_amdgcn_readlane(idxL, i);
                process_edge(s, idx, cur, first_seg, rx, ry, x, out, foff);
            }
        } else {
            for (int i = 0; i < cnt; ++i) {
                const int s   = __builtin_amdgcn_readlane(sL, i);
                const int idx = __builtin_amdgcn_readlane(idxL, i);
                process_edge(s, idx, cur, first_seg, rx, ry, x, out, foff);
            }
        }
        sL = sN; idxL = idxN;
    }

    // Last segment in range may straddle into the next wave's range -> atomic.
    float* dst = out + (size_t)cur * D_FEAT + foff;
    atomic_max_f32(dst + 0, rx);
    atomic_max_f32(dst + 1, ry);
}

// Pass 3: empty segments still hold -inf -> replace with 0 (reference's
// counts>0 select). Real data is finite, so -inf uniquely marks "untouched".
__global__ void __launch_bounds__(BLOCK_THREADS)
seg_fixup_kernel(uint32_t* __restrict__ out, int n4) {
    int i = blockIdx.x * blockDim.x + threadIdx.x;
    if (i < n4) {
        uint4 v = ((uint4*)out)[i];
        v.x = (v.x == NEG_INF_BITS) ? 0u : v.x;
        v.y = (v.y == NEG_INF_BITS) ? 0u : v.y;
        v.z = (v.z == NEG_INF_BITS) ? 0u : v.z;
        v.w = (v.w == NEG_INF_BITS) ? 0u : v.w;
        ((uint4*)out)[i] = v;
    }
}

extern "C" void kernel_launch(void* const* d_in, const int* in_sizes, int n_in,
                              void* d_out, int out_size, void* d_ws, size_t ws_size,
                              hipStream_t stream) {
    const float* x   = (const float*)d_in[0];   // [N, 64] f32
    const int*   nbr = (const int*)  d_in[1];   // [E] indices
    const int*   seg = (const int*)  d_in[2];   // [E] sorted segment ids
    float*       out = (float*)d_out;           // [N, 64] f32

    const int n_edges = in_sizes[1];
    const int n4      = out_size / 4;           // out_size = N*64, multiple of 4

    // Pass 1: seed -inf
    seg_init_kernel<<<(n4 + BLOCK_THREADS - 1) / BLOCK_THREADS,
                      BLOCK_THREADS, 0, stream>>>((uint32_t*)out, n4);

    // Pass 2: gather + sorted-segment max
    const int edges_per_wave = (n_edges + WAVES_TOTAL - 1) / WAVES_TOTAL;
    const int gather_blocks  = (WAVES_TOTAL * 32) / BLOCK_THREADS;
    seg_max_gather_kernel<<<gather_blocks, BLOCK_THREADS, 0, stream>>>(
        x, nbr, seg, out, n_edges, edges_per_wave);

    // Pass 3: -inf -> 0 for empty segments
    seg_fixup_kernel<<<(n4 + BLOCK_THREADS - 1) / BLOCK_THREADS,
                       BLOCK_THREADS, 0, stream>>>((uint32_t*)out, n4);
}